// VoxelFeatureEncoder_5583457485250
// MI455X (gfx1250) — compile-verified
//
#include <hip/hip_runtime.h>

typedef __attribute__((ext_vector_type(2))) float v2f;
typedef __attribute__((ext_vector_type(8))) float v8f;

#define NV 100000
#define TT 10
#define ROWS (NV * TT)          // 1,000,000 rows = 62,500 16-row tiles exactly
#define BN_EPS 1e-5f
#define WAVES_PER_BLK 4
#define GBLK ((ROWS / 16) / WAVES_PER_BLK)   // 15625 blocks, exact cover

#define SWZ_SWAPX16 0x401F      // ds_swizzle: xor_mask=0x10, and_mask=0x1F

__device__ __forceinline__ float swapx16(float x) {
  return __int_as_float(__builtin_amdgcn_ds_swizzle(__float_as_int(x), SWZ_SWAPX16));
}

// ---------------------------------------------------------------------------
// GEMM (ROWS x CIN) * (CIN x COUT) -> H (ROWS x COUT), plus per-block partial
// channel sums / sums-of-squares for BatchNorm statistics.
// One wave owns one 16-row tile and sweeps all COUT/16 column tiles with
// V_WMMA_F32_16X16X4_F32.  A-tile staged in LDS; K padded to a multiple of 4
// so the inner loop is branch-free; grid covers rows exactly => no guards.
//
// FUSE=true: the logical input row is rebuilt on the fly from the previous
// layer's pre-BN activations:  x[row] = mask ? [relu(h*sc+sh), pooled[voxel]]
// : 0  (CIN = 2*CPREV).  This avoids materializing the concat tensor.
// ---------------------------------------------------------------------------
template <int CIN, int COUT, bool FUSE>
__global__ __launch_bounds__(128) void gemm_stats_kernel(
    const float* __restrict__ X, const float* __restrict__ W,
    float* __restrict__ H, float* __restrict__ partials,
    const float* __restrict__ mask, const float* __restrict__ scaleIn,
    const float* __restrict__ shiftIn, const float* __restrict__ pooled) {
  constexpr int NT   = COUT / 16;
  constexpr int KP   = (CIN + 3) & ~3;       // padded K (8 for CIN=7)
  constexpr bool PAD = (KP != CIN);
  constexpr int CP   = CIN / 2;              // prev-layer channels (FUSE)
  constexpr int XS   = FUSE ? CP : CIN;      // row stride of X

  __shared__ __align__(16) float sX[WAVES_PER_BLK][16 * KP];
  __shared__ __align__(16) float sW[PAD ? KP * COUT : 1];  // zero-padded W
  __shared__ float sSum[WAVES_PER_BLK][COUT];
  __shared__ float sSq[WAVES_PER_BLK][COUT];

  const int tid  = threadIdx.x;
  const int lane = tid & 31;
  const int wave = tid >> 5;

  const int rowBase = (blockIdx.x * WAVES_PER_BLK + wave) * 16;

  if constexpr (PAD) {   // stage W zero-padded to KP rows (tiny: KP*COUT)
    for (int i = tid; i < KP * COUT; i += 128) {
      int k = i / COUT;
      sW[i] = (k < CIN) ? W[i] : 0.0f;       // W row-major: i = k*COUT + c
    }
  }

  // stage this wave's 16 x CIN A-tile into LDS
  if constexpr (FUSE) {
    for (int i = lane; i < 16 * CIN; i += 32) {
      int r = i / CIN, c = i - r * CIN;      // CIN power of two -> shifts
      int row = rowBase + r;
      int v = row / TT;
      float m = mask[row];
      float val;
      if (c < CP) {
        float h = X[(size_t)row * CP + c];
        val = (m > 0.f) ? fmaxf(h * scaleIn[c] + shiftIn[c], 0.f) : 0.f;
      } else {
        val = (m > 0.f) ? pooled[(size_t)v * CP + (c - CP)] : 0.f;
      }
      sX[wave][i] = val;
    }
  } else if constexpr (PAD) {
    for (int i = lane; i < 16 * KP; i += 32) {
      int r = i / KP, c = i - r * KP;        // KP power of two -> shifts
      sX[wave][i] = (c < CIN) ? X[(size_t)(rowBase + r) * CIN + c] : 0.0f;
    }
  } else {
    for (int i = lane; i < 16 * CIN; i += 32) {
      int r = i / CIN, c = i - r * CIN;
      sX[wave][i] = X[(size_t)(rowBase + r) * CIN + c];
    }
  }
  // warm L2 for the next block's tile (hint only: global_prefetch_b8)
  if (rowBase + WAVES_PER_BLK * 16 + 16 <= ROWS)
    __builtin_prefetch(
        X + ((size_t)rowBase + WAVES_PER_BLK * 16) * XS + lane * (XS / 2 + 1),
        0, 1);
  __syncthreads();

  const int arow  = lane & 15;        // A: M index
  const int koff  = (lane >> 4) * 2;  // A/B: K-pair selected by lane half
  const int bcol  = lane & 15;        // B/D: N index
  const int rhalf = (lane >> 4) * 8;

  // per-lane output base: the 8 stores below get immediate offsets
  float* __restrict__ Hp = H + (size_t)(rowBase + rhalf) * COUT + bcol;

#pragma unroll
  for (int nt = 0; nt < NT; ++nt) {
    v8f acc = {0.f, 0.f, 0.f, 0.f, 0.f, 0.f, 0.f, 0.f};
#pragma unroll
    for (int k = 0; k < KP; k += 4) {
      const int kk = k + koff;
      v2f a = *(const v2f*)&sX[wave][arow * KP + kk];   // ds_load_b64
      v2f b;
      const int col = nt * 16 + bcol;
      if constexpr (PAD) {
        b.x = sW[kk * COUT + col];
        b.y = sW[(kk + 1) * COUT + col];
      } else {
        b.x = W[kk * COUT + col];
        b.y = W[(kk + 1) * COUT + col];
      }
      acc = __builtin_amdgcn_wmma_f32_16x16x4_f32(
          false, a, false, b, (short)0, acc, false, false);
    }
    // store D tile (no guards: grid covers rows exactly) + fold stats
    float s = 0.f, q = 0.f;
#pragma unroll
    for (int r = 0; r < 8; ++r) {
      float v = acc[r];
      Hp[(size_t)r * COUT + nt * 16] = v;
      s += v;
      q += v * v;
    }
    s += swapx16(s);   // fold the two 8-row halves (ds_swizzle SWAPX16)
    q += swapx16(q);
    if (lane < 16) {
      sSum[wave][nt * 16 + lane] = s;
      sSq[wave][nt * 16 + lane]  = q;
    }
  }
  __syncthreads();
  if (tid < COUT) {   // one block partial per channel (deterministic order)
    float s = sSum[0][tid] + sSum[1][tid] + sSum[2][tid] + sSum[3][tid];
    float q = sSq[0][tid] + sSq[1][tid] + sSq[2][tid] + sSq[3][tid];
    partials[(size_t)blockIdx.x * (2 * COUT) + tid]        = s;
    partials[(size_t)blockIdx.x * (2 * COUT) + COUT + tid] = q;
  }
}

// ---------------------------------------------------------------------------
// Reduce block partials -> BN scale/shift per channel.  One block per channel,
// strided accumulation + fixed LDS tree => bitwise deterministic.
// ---------------------------------------------------------------------------
__global__ __launch_bounds__(256) void reduce_stats_kernel(
    const float* __restrict__ partials, int nBlocks, int C,
    const float* __restrict__ gamma, const float* __restrict__ beta,
    float* __restrict__ scale, float* __restrict__ shift) {
  __shared__ float ls[256], lq[256];
  const int c = blockIdx.x;
  float s = 0.f, q = 0.f;
  for (int b = threadIdx.x; b < nBlocks; b += 256) {
    s += partials[(size_t)b * 2 * C + c];
    q += partials[(size_t)b * 2 * C + C + c];
  }
  ls[threadIdx.x] = s;
  lq[threadIdx.x] = q;
  __syncthreads();
  for (int off = 128; off > 0; off >>= 1) {
    if (threadIdx.x < off) {
      ls[threadIdx.x] += ls[threadIdx.x + off];
      lq[threadIdx.x] += lq[threadIdx.x + off];
    }
    __syncthreads();
  }
  if (threadIdx.x == 0) {
    const float invR = 1.0f / (float)ROWS;
    float mean = ls[0] * invR;
    float var  = lq[0] * invR - mean * mean;   // biased, as torch/jax use
    float sc   = gamma[c] * rsqrtf(var + BN_EPS);
    scale[c] = sc;
    shift[c] = beta[c] - mean * sc;
  }
}

// ---------------------------------------------------------------------------
// Per-voxel max pool of relu(bn(h))*mask over T -> pooled (NV x C).
// float4-vectorized.  ReLU >= 0 and >=1 valid point per voxel => init 0.
// ---------------------------------------------------------------------------
template <int C>
__global__ __launch_bounds__(256) void bn_pool_kernel(
    const float* __restrict__ H, const float* __restrict__ mask,
    const float* __restrict__ scale, const float* __restrict__ shift,
    float* __restrict__ pooled) {
  constexpr int C4 = C / 4;
  const int g = blockIdx.x * 256 + threadIdx.x;
  if (g >= NV * C4) return;
  const int v = g / C4, c4 = g - v * C4;
  const float4 sc = ((const float4*)scale)[c4];
  const float4 sh = ((const float4*)shift)[c4];
  const float4* __restrict__ H4 = (const float4*)H;

  float4 best = make_float4(0.f, 0.f, 0.f, 0.f);
#pragma unroll
  for (int t = 0; t < TT; ++t) {
    float m = mask[v * TT + t];
    float4 h = H4[(size_t)(v * TT + t) * C4 + c4];
    if (m > 0.f) {
      best.x = fmaxf(best.x, fmaxf(h.x * sc.x + sh.x, 0.f));
      best.y = fmaxf(best.y, fmaxf(h.y * sc.y + sh.y, 0.f));
      best.z = fmaxf(best.z, fmaxf(h.z * sc.z + sh.z, 0.f));
      best.w = fmaxf(best.w, fmaxf(h.w * sc.w + sh.w, 0.f));
    }
  }
  ((float4*)pooled)[(size_t)v * C4 + c4] = best;
}

// ---------------------------------------------------------------------------
extern "C" void kernel_launch(void* const* d_in, const int* in_sizes, int n_in,
                              void* d_out, int out_size, void* d_ws,
                              size_t ws_size, hipStream_t stream) {
  const float* pts  = (const float*)d_in[0];
  const float* mask = (const float*)d_in[1];
  const float* W1   = (const float*)d_in[2];
  const float* g1   = (const float*)d_in[3];
  const float* b1   = (const float*)d_in[4];
  const float* W2   = (const float*)d_in[5];
  const float* g2   = (const float*)d_in[6];
  const float* b2   = (const float*)d_in[7];
  const float* Wf   = (const float*)d_in[8];
  const float* gf   = (const float*)d_in[9];
  const float* bf   = (const float*)d_in[10];
  float* out = (float*)d_out;

  // workspace (floats): h1 | h2 | h3 | partials | pooled | scale | shift
  float* h1       = (float*)d_ws;                       // ROWS*32
  float* h2       = h1 + (size_t)ROWS * 32;             // ROWS*64
  float* h3       = h2 + (size_t)ROWS * 64;             // ROWS*128
  float* partials = h3 + (size_t)ROWS * 128;            // GBLK*256
  float* pooled   = partials + (size_t)GBLK * 256;      // NV*64 (max)
  float* scale    = pooled + (size_t)NV * 64;           // 128
  float* shift    = scale + 128;                        // 128

  // ---- layer 1: (1M x 7) @ (7 x 32) ----
  gemm_stats_kernel<7, 32, false><<<GBLK, 128, 0, stream>>>(
      pts, W1, h1, partials, nullptr, nullptr, nullptr, nullptr);
  reduce_stats_kernel<<<32, 256, 0, stream>>>(partials, GBLK, 32, g1, b1,
                                              scale, shift);
  bn_pool_kernel<32><<<(NV * 8 + 255) / 256, 256, 0, stream>>>(
      h1, mask, scale, shift, pooled);

  // ---- layer 2: (1M x 64) @ (64 x 64), input rebuilt from h1 ----
  gemm_stats_kernel<64, 64, true><<<GBLK, 128, 0, stream>>>(
      h1, W2, h2, partials, mask, scale, shift, pooled);
  reduce_stats_kernel<<<64, 256, 0, stream>>>(partials, GBLK, 64, g2, b2,
                                              scale, shift);
  bn_pool_kernel<64><<<(NV * 16 + 255) / 256, 256, 0, stream>>>(
      h2, mask, scale, shift, pooled);

  // ---- final: (1M x 128) @ (128 x 128), input rebuilt from h2 ----
  gemm_stats_kernel<128, 128, true><<<GBLK, 128, 0, stream>>>(
      h2, Wf, h3, partials, mask, scale, shift, pooled);
  reduce_stats_kernel<<<128, 256, 0, stream>>>(partials, GBLK, 128, gf, bf,
                                               scale, shift);
  // output = per-voxel max of relu(bn(h3))*mask  (same pooling pattern)
  bn_pool_kernel<128><<<(NV * 32 + 255) / 256, 256, 0, stream>>>(
      h3, mask, scale, shift, out);
}